// CausalSelfAttention_6038724018433
// MI455X (gfx1250) — compile-verified
//
#include <hip/hip_runtime.h>

// CDNA5 / gfx1250 causal self-attention, fp32 end-to-end via V_WMMA_F32_16X16X4_F32.
// Phase 1: qkv = x @ W + b, 64x64 tile per wave (16 independent WMMA accumulator
//          chains), Q/K/V scattered head-major [B,NH,T,HS] into workspace.
// Phase 2: streaming (flash) causal attention, one wave per 16-row query tile,
//          dual-chain S accumulation + interleaved O-chunk accumulation.

typedef float v2f __attribute__((ext_vector_type(2)));
typedef float v8f __attribute__((ext_vector_type(8)));

#define B_   2
#define T_   2048
#define C_   768
#define NH_  12
#define HS_  64
#define N3_  (3 * C_)

static __device__ __forceinline__ v8f wmma4(v2f a, v2f b, v8f c) {
    // D = A(16x4,f32) * B(4x16,f32) + C(16x16,f32)
    return __builtin_amdgcn_wmma_f32_16x16x4_f32(false, a, false, b, (short)0, c,
                                                 false, false);
}

// ---------------------------------------------------------------------------
// Kernel 1: QKV projection. One wave computes a 64x64 tile of [4096 x 2304].
//   A-frag: lane(0..15)=row M, half-wave selects K pair {2h, 2h+1}.
//   B-frag: lane(0..15)=col N, half-wave selects K pair {2h, 2h+1}.
//   C/D:    lane&15 = col, VGPR r -> row r + 8*(lane>>4).
// Per K-step: 4 b64 A-loads + 8 b32 B-loads feed 16 independent WMMAs.
// ---------------------------------------------------------------------------
__global__ __launch_bounds__(32) void qkv_gemm_k(const float* __restrict__ x,
                                                 const float* __restrict__ w,
                                                 const float* __restrict__ bias,
                                                 float* __restrict__ Qp,
                                                 float* __restrict__ Kp,
                                                 float* __restrict__ Vp) {
    const int lane = threadIdx.x;
    const int l16  = lane & 15;
    const int half = lane >> 4;
    const int n0   = blockIdx.x * 64;  // column tile origin within 3C
    const int m0   = blockIdx.y * 64;  // row tile origin within B*T

    const float* xRow[4];
    const float* wCol[4];
#pragma unroll
    for (int m = 0; m < 4; ++m) xRow[m] = x + (size_t)(m0 + 16 * m + l16) * C_;
#pragma unroll
    for (int n = 0; n < 4; ++n) wCol[n] = w + n0 + 16 * n + l16;

    v8f acc[4][4];
#pragma unroll
    for (int m = 0; m < 4; ++m)
#pragma unroll
        for (int n = 0; n < 4; ++n)
#pragma unroll
            for (int r = 0; r < 8; ++r) acc[m][n][r] = 0.0f;

    for (int k = 0; k < C_ / 4; ++k) {
        const int kk = 4 * k + 2 * half;
        v2f a[4], b[4];
#pragma unroll
        for (int m = 0; m < 4; ++m) a[m] = *(const v2f*)(xRow[m] + kk);
#pragma unroll
        for (int n = 0; n < 4; ++n) {
            b[n].x = wCol[n][(size_t)kk * N3_];
            b[n].y = wCol[n][(size_t)(kk + 1) * N3_];
        }
#pragma unroll
        for (int m = 0; m < 4; ++m)
#pragma unroll
            for (int n = 0; n < 4; ++n)
                acc[m][n] = wmma4(a[m], b[n], acc[m][n]);
    }

    // Epilogue: fused bias add + head-major scatter into Q/K/V.
#pragma unroll
    for (int n = 0; n < 4; ++n) {
        const int   col   = n0 + 16 * n + l16;
        const float bv    = bias[col];
        const int   which = col / C_;              // 0=q, 1=k, 2=v (uniform per tile)
        const int   rem   = col - which * C_;
        const int   h     = rem / HS_;
        const int   d     = rem - h * HS_;
        float* dst = (which == 0) ? Qp : (which == 1) ? Kp : Vp;
#pragma unroll
        for (int m = 0; m < 4; ++m)
#pragma unroll
            for (int r = 0; r < 8; ++r) {
                const int row = m0 + 16 * m + r + 8 * half;
                const int bb  = row >> 11;          // / T_
                const int t   = row & (T_ - 1);     // % T_
                dst[(((size_t)bb * NH_ + h) * T_ + t) * HS_ + d] = acc[m][n][r] + bv;
            }
    }
}

// ---------------------------------------------------------------------------
// Kernel 2: streaming causal attention. One wave per (b*NH+h, 16-row q tile).
// ---------------------------------------------------------------------------
__global__ __launch_bounds__(32) void flash_attn_k(const float* __restrict__ Qp,
                                                   const float* __restrict__ Kp,
                                                   const float* __restrict__ Vp,
                                                   float* __restrict__ out) {
    __shared__ float pLds[16 * 16];  // P tile bounce buffer (C-layout -> A-layout)

    const int lane = threadIdx.x;
    const int l16  = lane & 15;
    const int half = lane >> 4;
    const int qb   = blockIdx.x;       // query tile index (16 rows)
    const int bh   = blockIdx.y;       // fused (b, head)
    const float scale = 0.125f;        // 1/sqrt(64)

    // Preload Q as 16 A-frags covering K = 0..63.
    const float* qRow = Qp + ((size_t)bh * T_ + qb * 16 + l16) * HS_;
    v2f qa[16];
#pragma unroll
    for (int i = 0; i < 16; ++i) qa[i] = *(const v2f*)(qRow + 4 * i + 2 * half);

    v8f o[4];
#pragma unroll
    for (int cc = 0; cc < 4; ++cc)
#pragma unroll
        for (int r = 0; r < 8; ++r) o[cc][r] = 0.0f;

    float rowmax[8], rowsum[8];
#pragma unroll
    for (int r = 0; r < 8; ++r) { rowmax[r] = -1e30f; rowsum[r] = 0.0f; }

    for (int j = 0; j <= qb; ++j) {    // wave-uniform causal key-block loop
        const float* kRow  = Kp + ((size_t)bh * T_ + j * 16 + l16) * HS_;
        const float* vBase = Vp + ((size_t)bh * T_ + j * 16) * HS_;

        // Prefetch next block of K/V (stays inside workspace; gfx1250
        // global_prefetch_b8 path).
        __builtin_prefetch(kRow + 16 * HS_, 0, 3);
        __builtin_prefetch(vBase + 16 * HS_ + l16, 0, 3);

        // ---- S = Q K^T, two independent accumulator chains ----
        v8f s0, s1;
#pragma unroll
        for (int r = 0; r < 8; ++r) { s0[r] = 0.0f; s1[r] = 0.0f; }
#pragma unroll
        for (int i = 0; i < 8; ++i) {
            v2f kb0 = *(const v2f*)(kRow + 8 * i + 2 * half);        // frag 2i
            v2f kb1 = *(const v2f*)(kRow + 8 * i + 4 + 2 * half);    // frag 2i+1
            s0 = wmma4(qa[2 * i],     kb0, s0);
            s1 = wmma4(qa[2 * i + 1], kb1, s1);
        }
        v8f s = s0 + s1;

        // ---- scale + causal mask (only bites on the diagonal block) ----
        const int ncol = j * 16 + l16;
#pragma unroll
        for (int r = 0; r < 8; ++r) {
            const int mrow = qb * 16 + r + 8 * half;
            const float v  = s[r] * scale;
            s[r] = (ncol <= mrow) ? v : -1e30f;   // v_cndmask, EXEC untouched
        }

        // ---- online softmax: each row lives in one 16-lane half ----
        float alpha[8];
#pragma unroll
        for (int r = 0; r < 8; ++r) {
            float m = s[r];
#pragma unroll
            for (int off = 1; off < 16; off <<= 1)
                m = fmaxf(m, __shfl_xor(m, off, 32));
            const float mnew = fmaxf(rowmax[r], m);
            alpha[r]  = __expf(rowmax[r] - mnew);
            rowmax[r] = mnew;
            const float p = __expf(s[r] - mnew);
            s[r] = p;
            float sum = p;
#pragma unroll
            for (int off = 1; off < 16; off <<= 1)
                sum += __shfl_xor(sum, off, 32);
            rowsum[r] = rowsum[r] * alpha[r] + sum;
        }
#pragma unroll
        for (int cc = 0; cc < 4; ++cc)
#pragma unroll
            for (int r = 0; r < 8; ++r) o[cc][r] *= alpha[r];

        // ---- P: C-layout -> A-layout through LDS (DS is in-order per wave) ----
#pragma unroll
        for (int r = 0; r < 8; ++r)
            pLds[(r + 8 * half) * 16 + l16] = s[r];

        // ---- O += P V: i outer / chunk inner so consecutive WMMAs hit
        //      different accumulators (independent chains) ----
#pragma unroll
        for (int i = 0; i < 4; ++i) {
            const int kk = 4 * i + 2 * half;
            v2f pa = *(const v2f*)(&pLds[l16 * 16 + kk]);
#pragma unroll
            for (int cc = 0; cc < 4; ++cc) {
                v2f bv;
                bv.x = vBase[(size_t)kk * HS_ + cc * 16 + l16];
                bv.y = vBase[(size_t)(kk + 1) * HS_ + cc * 16 + l16];
                o[cc] = wmma4(pa, bv, o[cc]);
            }
        }
    }

    // ---- finalize: divide by row sums, store [B,NH,T,HS] ----
#pragma unroll
    for (int cc = 0; cc < 4; ++cc)
#pragma unroll
        for (int r = 0; r < 8; ++r) {
            const int row = qb * 16 + r + 8 * half;
            out[((size_t)bh * T_ + row) * HS_ + cc * 16 + l16] =
                o[cc][r] / rowsum[r];
        }
}

// ---------------------------------------------------------------------------
extern "C" void kernel_launch(void* const* d_in, const int* in_sizes, int n_in,
                              void* d_out, int out_size, void* d_ws, size_t ws_size,
                              hipStream_t stream) {
    (void)in_sizes; (void)n_in; (void)out_size; (void)ws_size;
    const float* x    = (const float*)d_in[0];   // [B,T,C]
    const float* w    = (const float*)d_in[1];   // [C,3C]
    const float* bias = (const float*)d_in[2];   // [3C]
    float* out = (float*)d_out;                  // [B,NH,T,HS]

    const size_t n = (size_t)B_ * NH_ * T_ * HS_;
    float* Qp = (float*)d_ws;
    float* Kp = Qp + n;
    float* Vp = Kp + n;

    dim3 g1(N3_ / 64, (B_ * T_) / 64);           // 36 x 64 waves, 64x64 tiles
    qkv_gemm_k<<<g1, 32, 0, stream>>>(x, w, bias, Qp, Kp, Vp);

    dim3 g2(T_ / 16, B_ * NH_);                  // 128 x 24 waves
    flash_attn_k<<<g2, 32, 0, stream>>>(Qp, Kp, Vp, out);
}